// MistralFlashAttention2_79035988181513
// MI455X (gfx1250) — compile-verified
//
#include <hip/hip_runtime.h>

// ---- problem constants (match reference) ----
#define HID    4096
#define NH     32
#define KVHN   8
#define DH     128
#define BB     4
#define QQ     256
#define KVLEN  2048
#define HIST   1792
#define BSZ    64
#define NBLK   32            // blocks per sequence = KV/BS
#define WIN    1024
#define NEGINF (-1.0e30f)

typedef __attribute__((ext_vector_type(2))) float v2f;
typedef __attribute__((ext_vector_type(8))) float v8f;

static __device__ __forceinline__ v8f zero8() {
  v8f z = {0.f,0.f,0.f,0.f,0.f,0.f,0.f,0.f};
  return z;
}

// D = A(16x4 f32) * B(4x16 f32) + C(16x16 f32)  -> v_wmma_f32_16x16x4_f32
static __device__ __forceinline__ v8f wmma4(v2f a, v2f b, v8f c) {
  return __builtin_amdgcn_wmma_f32_16x16x4_f32(false, a, false, b, (short)0, c,
                                               false, false);
}

// CDNA5 async global->LDS copy (16B per lane), tracked by ASYNCcnt.
// LDS operand is the wave-relative LDS offset = low 32 bits of the generic ptr.
static __device__ __forceinline__ void async_g2l_b128(void* lds_ptr,
                                                      const void* gptr) {
  unsigned loff = (unsigned)(unsigned long long)lds_ptr;
  unsigned long long g = (unsigned long long)gptr;
  asm volatile("global_load_async_to_lds_b128 %0, %1, off"
               :: "v"(loff), "v"(g) : "memory");
}
static __device__ __forceinline__ void wait_async() {
  asm volatile("s_wait_asynccnt 0x0" ::: "memory");
}

// ---------------------------------------------------------------------------
// Generic fp32 WMMA GEMM: C[M,N] = A[M,K] @ B[K,N]
// B is selected per 128-col tile among {B0,B1,B2} with column boundaries
// nb1, nb2 (cumulative), so the QKV projection is a single fused launch.
// Block = 256 threads = 8 waves (2x4), block tile 128x128, wave tile 64x32.
// Tiles staged with global_load_async_to_lds_b128.
// ---------------------------------------------------------------------------
__global__ __launch_bounds__(256) void gemm_f32_wmma(
    const float* __restrict__ A, int lda,
    const float* __restrict__ B0, const float* __restrict__ B1,
    const float* __restrict__ B2,
    int nb1, int nb2, int ldb0, int ldb1, int ldb2,
    float* __restrict__ C, int ldc, int K)
{
  __shared__ float Alds[128][20];   // 128 rows x 16 k (+4 pad: bank stride 20)
  __shared__ float Blds[16][132];   // 16 k x 128 n  (+4 pad)

  const int tid  = threadIdx.x;
  const int lane = tid & 31;
  const int wv   = tid >> 5;
  const int wm0  = (wv >> 2) * 64;      // wave row base in block tile
  const int wn0  = (wv & 3) * 32;       // wave col base in block tile
  const int m0   = blockIdx.y * 128;
  const int n0   = blockIdx.x * 128;
  const int lh   = lane >> 4;           // K-half select for fragments
  const int ll   = lane & 15;

  const float* Bp; int ldb; int nc;
  if (n0 < nb1)      { Bp = B0; ldb = ldb0; nc = n0;       }
  else if (n0 < nb2) { Bp = B1; ldb = ldb1; nc = n0 - nb1; }
  else               { Bp = B2; ldb = ldb2; nc = n0 - nb2; }

  v8f acc[4][2];
  for (int i = 0; i < 4; ++i)
    for (int j = 0; j < 2; ++j) acc[i][j] = zero8();

  for (int k0 = 0; k0 < K; k0 += 16) {
    __syncthreads();
    // stage A tile 128x16 (512 x b128, 2 per thread) via async DMA to LDS
    for (int j = 0; j < 2; ++j) {
      int idx = tid * 2 + j;
      int row = idx >> 2;
      int kk4 = (idx & 3) * 4;
      async_g2l_b128(&Alds[row][kk4],
                     &A[(size_t)(m0 + row) * lda + k0 + kk4]);
    }
    // stage B tile 16x128 (512 x b128, 2 per thread)
    for (int j = 0; j < 2; ++j) {
      int idx = tid * 2 + j;
      int row = idx >> 5;
      int n4  = (idx & 31) * 4;
      async_g2l_b128(&Blds[row][n4],
                     &Bp[(size_t)(k0 + row) * ldb + nc + n4]);
    }
    wait_async();
    __syncthreads();

    for (int kk = 0; kk < 16; kk += 4) {
      v2f afrag[4];
      for (int tm = 0; tm < 4; ++tm) {
        int row = wm0 + tm * 16 + ll;               // M = lane 0..15
        afrag[tm] = *(const v2f*)&Alds[row][kk + lh * 2];  // K half per lane-half
      }
      v2f bfrag[2];
      for (int tn = 0; tn < 2; ++tn) {
        int col = wn0 + tn * 16 + ll;               // N = lane 0..15
        v2f b;
        b.x = Blds[kk + lh * 2 + 0][col];
        b.y = Blds[kk + lh * 2 + 1][col];
        bfrag[tn] = b;
      }
      for (int tm = 0; tm < 4; ++tm)
        for (int tn = 0; tn < 2; ++tn)
          acc[tm][tn] = wmma4(afrag[tm], bfrag[tn], acc[tm][tn]);
    }
  }

  // store: C layout -> vgpr r holds rows M=r (lanes 0-15) / M=r+8 (lanes 16-31)
  for (int tm = 0; tm < 4; ++tm)
    for (int tn = 0; tn < 2; ++tn)
      for (int r = 0; r < 8; ++r) {
        int row = m0 + wm0 + tm * 16 + r + lh * 8;
        int col = n0 + wn0 + tn * 16 + ll;
        C[(size_t)row * ldc + col] = acc[tm][tn][r];
      }
}

// ---------------------------------------------------------------------------
// RoPE in place on the fused QKV rows: row t = [ q(4096) | k(1024) | v(1024) ]
// ---------------------------------------------------------------------------
__global__ __launch_bounds__(256) void rope_kernel(
    float* __restrict__ qkv, const int* __restrict__ pos_ids)
{
  const int t   = blockIdx.x;
  const int pos = pos_ids[t];
  float* base   = qkv + (size_t)t * (HID + 2 * KVHN * DH);
  const int npairs = (NH + KVHN) * 64;
  for (int i = threadIdx.x; i < npairs; i += blockDim.x) {
    float* p; int d;
    if (i < NH * 64) {
      int head = i >> 6; d = i & 63;
      p = base + head * DH;
    } else {
      int j = i - NH * 64;
      int head = j >> 6; d = j & 63;
      p = base + HID + head * DH;
    }
    float inv = powf(10000.0f, -(float)d * (1.0f / 64.0f));
    float f   = (float)pos * inv;
    float c = cosf(f), s = sinf(f);
    float x1 = p[d], x2 = p[d + 64];
    p[d]      = x1 * c - x2 * s;
    p[d + 64] = x2 * c + x1 * s;
  }
}

// ---------------------------------------------------------------------------
// Flash attention with online softmax, sliding window + causal mask.
// Grid: x = b*32+head (128), y = q-block of 64 rows (4). Block = 128 thr (4 waves).
// Wave w owns q rows [qb0+16w, qb0+16w+16). q fragments (scaled) live in VGPRs
// across the whole KV loop. KV tiles of 64 never straddle HIST, so history
// tiles come from the immutable cache inputs, fresh tiles from workspace QKV.
// K/V tiles staged with global_load_async_to_lds_b128.
// ---------------------------------------------------------------------------
__global__ __launch_bounds__(128) void attn_kernel(
    const float* __restrict__ qkv,
    const float* __restrict__ k_cache,
    const float* __restrict__ v_cache,
    const int*   __restrict__ block_offsets,
    float* __restrict__ ows)
{
  __shared__ float Klds[64][132];
  __shared__ float Vlds[64][132];
  __shared__ float Plds[4][16][68];

  const int ROW = HID + 2 * KVHN * DH;     // 6144
  const int bh  = blockIdx.x;
  const int b   = bh >> 5;
  const int hh  = bh & 31;
  const int kvh = hh >> 2;                 // GQA: 4 q-heads per kv-head
  const int qb0 = blockIdx.y * 64;
  const int tid = threadIdx.x;
  const int lane = tid & 31;
  const int w   = tid >> 5;
  const int lh  = lane >> 4;
  const int ll  = lane & 15;

  // preload q fragments for this wave's 16 rows, folded softmax scale
  const float scale = 0.088388347648318447f;   // 1/sqrt(128)
  v2f qa[32];
  {
    int srow = qb0 + w * 16 + ll;
    const float* qp = qkv + (size_t)(b * QQ + srow) * ROW + hh * DH;
    for (int s = 0; s < 32; ++s) {
      v2f a = *(const v2f*)&qp[s * 4 + lh * 2];
      a.x *= scale; a.y *= scale;
      qa[s] = a;
    }
  }

  v8f oacc[8];
  for (int i = 0; i < 8; ++i) oacc[i] = zero8();
  float m_run[8], l_run[8];
  for (int r = 0; r < 8; ++r) { m_run[r] = NEGINF; l_run[r] = 0.f; }

  const int kt_lo = (HIST + qb0 - (WIN - 1)) >> 6;
  const int kt_hi = (HIST + qb0 + 63) >> 6;

  for (int kt = kt_lo; kt <= kt_hi; ++kt) {
    const int kv0 = kt << 6;
    __syncthreads();
    // ---- stage K/V tile (64 x 128 each) via async DMA ----
    if (kv0 >= HIST) {
      const float* kb = qkv + (size_t)(b * QQ + (kv0 - HIST)) * ROW + HID + kvh * DH;
      const float* vb = kb + KVHN * DH;
      for (int j = 0; j < 16; ++j) {
        int idx = tid + j * 128;
        int row = idx >> 5;
        int d4  = (idx & 31) * 4;
        async_g2l_b128(&Klds[row][d4], &kb[(size_t)row * ROW + d4]);
        async_g2l_b128(&Vlds[row][d4], &vb[(size_t)row * ROW + d4]);
      }
    } else {
      const int blk = block_offsets[b * NBLK + kt];
      const float* kb = k_cache + ((size_t)blk * BSZ * KVHN + kvh) * DH;
      const float* vb = v_cache + ((size_t)blk * BSZ * KVHN + kvh) * DH;
      for (int j = 0; j < 16; ++j) {
        int idx = tid + j * 128;
        int row = idx >> 5;
        int d4  = (idx & 31) * 4;
        async_g2l_b128(&Klds[row][d4], &kb[(size_t)row * KVHN * DH + d4]);
        async_g2l_b128(&Vlds[row][d4], &vb[(size_t)row * KVHN * DH + d4]);
      }
    }
    wait_async();
    __syncthreads();

    // ---- scores S = q @ K^T (16 x 64 per wave) ----
    v8f sacc[4];
    for (int n = 0; n < 4; ++n) sacc[n] = zero8();
    for (int s = 0; s < 32; ++s) {
      int kk = s * 4 + lh * 2;
      for (int n = 0; n < 4; ++n) {
        int tok = n * 16 + ll;
        v2f bfr; bfr.x = Klds[tok][kk]; bfr.y = Klds[tok][kk + 1];
        sacc[n] = wmma4(qa[s], bfr, sacc[n]);
      }
    }
    // ---- causal + sliding-window mask ----
    for (int n = 0; n < 4; ++n) {
      int kpos = kv0 + n * 16 + ll;
      for (int r = 0; r < 8; ++r) {
        int qpos = HIST + qb0 + w * 16 + r + lh * 8;
        bool valid = (kpos <= qpos) && (kpos > qpos - WIN);
        if (!valid) sacc[n][r] = NEGINF;
      }
    }
    // ---- online softmax update (row-wise across the 16 lanes of a half) ----
    float alpha[8];
    for (int r = 0; r < 8; ++r) {
      float rmax = sacc[0][r];
      rmax = fmaxf(rmax, sacc[1][r]);
      rmax = fmaxf(rmax, sacc[2][r]);
      rmax = fmaxf(rmax, sacc[3][r]);
      for (int off = 1; off <= 8; off <<= 1)
        rmax = fmaxf(rmax, __shfl_xor(rmax, off, 32));
      float mo = m_run[r];
      float mn = fmaxf(mo, rmax);
      float a  = __expf(mo - mn);        // finite sentinel -> never NaN
      float rsum = 0.f;
      for (int n = 0; n < 4; ++n) {
        float p = __expf(sacc[n][r] - mn);
        sacc[n][r] = p;
        rsum += p;
      }
      for (int off = 1; off <= 8; off <<= 1)
        rsum += __shfl_xor(rsum, off, 32);
      l_run[r] = a * l_run[r] + rsum;
      m_run[r] = mn;
      alpha[r] = a;
    }
    for (int tn = 0; tn < 8; ++tn)
      for (int r = 0; r < 8; ++r)
        oacc[tn][r] *= alpha[r];
    // ---- re-fragment P (C-layout -> A-layout) via per-wave LDS slice ----
    for (int n = 0; n < 4; ++n)
      for (int r = 0; r < 8; ++r)
        Plds[w][r + lh * 8][n * 16 + ll] = sacc[n][r];
    // ---- O += P @ V ----
    for (int s = 0; s < 16; ++s) {
      int kk = s * 4 + lh * 2;
      v2f pa = *(const v2f*)&Plds[w][ll][kk];
      for (int tn = 0; tn < 8; ++tn) {
        int col = tn * 16 + ll;
        v2f bfr; bfr.x = Vlds[kk][col]; bfr.y = Vlds[kk + 1][col];
        oacc[tn] = wmma4(pa, bfr, oacc[tn]);
      }
    }
  }

  // ---- normalize and store O [t, hh, d] ----
  for (int r = 0; r < 8; ++r) {
    float invl = 1.0f / l_run[r];
    int t = b * QQ + qb0 + w * 16 + r + lh * 8;
    float* orow = ows + (size_t)t * (NH * DH) + hh * DH;
    for (int tn = 0; tn < 8; ++tn)
      orow[tn * 16 + ll] = oacc[tn][r] * invl;
  }
}

// ---------------------------------------------------------------------------
extern "C" void kernel_launch(void* const* d_in, const int* in_sizes, int n_in,
                              void* d_out, int out_size, void* d_ws, size_t ws_size,
                              hipStream_t stream) {
  const float* hs = (const float*)d_in[0];
  const float* wq = (const float*)d_in[1];
  const float* wk = (const float*)d_in[2];
  const float* wv = (const float*)d_in[3];
  const float* wo = (const float*)d_in[4];
  const float* kc = (const float*)d_in[5];
  const float* vc = (const float*)d_in[6];
  const int*   pos = (const int*)d_in[7];
  const int*   boff = (const int*)d_in[11];

  const int T = BB * QQ;                      // 1024
  const int ROW = HID + 2 * KVHN * DH;        // 6144
  float* qkv = (float*)d_ws;                                    // [1024, 6144]
  float* ows = (float*)((char*)d_ws + (size_t)T * ROW * sizeof(float)); // [1024, 4096]
  float* out = (float*)d_out;

  // 1) fused QKV projection: [1024,4096] @ [4096, 4096|1024|1024]
  gemm_f32_wmma<<<dim3(ROW / 128, T / 128), 256, 0, stream>>>(
      hs, HID, wq, wk, wv, HID, HID + KVHN * DH, HID, KVHN * DH, KVHN * DH,
      qkv, ROW, HID);

  // 2) RoPE in place on q and k parts
  rope_kernel<<<T, 256, 0, stream>>>(qkv, pos);

  // 3) paged flash attention (history from cache inputs, fresh from qkv)
  attn_kernel<<<dim3(BB * NH, QQ / 64), 128, 0, stream>>>(qkv, kc, vc, boff, ows);

  // 4) output projection: [1024,4096] @ [4096,4096]
  gemm_f32_wmma<<<dim3(HID / 128, T / 128), 256, 0, stream>>>(
      ows, NH * DH, wo, wo, wo, HID, 2 * HID, HID, HID, HID,
      out, HID, HID);
}